// WHVILinear_41910290874902
// MI455X (gfx1250) — compile-verified
//
#include <hip/hip_runtime.h>
#include <hip/hip_bf16.h>

// ---------------------------------------------------------------------------
// WHVI linear layer, y = s1 ⊙ FWHT( u ⊙ FWHT( s2 ⊙ x ) ), u = mu + softplus(rho)*eps
// FWHT_4096 = H16 ⊗ H16 ⊗ H16  ->  three passes of 16x16 Hadamard matmuls,
// each realized as a chain of four V_WMMA_F32_16X16X4_F32 (full fp32).
// Memory-bound design: 256 MB total HBM traffic, ~11 us floor @ 23.3 TB/s.
// ---------------------------------------------------------------------------

#define WHVI_D 4096
#define WHVI_WAVES 4   // waves per block, each wave owns one row in LDS

typedef __attribute__((ext_vector_type(2))) float v2f;
typedef __attribute__((ext_vector_type(8))) float v8f;

// H16[m][k] entry including the per-stage 1/sqrt(16) scale: +-0.25
__device__ __forceinline__ float whvi_h16(int m, int k) {
    return (__popc(m & k) & 1) ? -0.25f : 0.25f;
}

// One radix-16 FWHT pass over a 4096-float wave-private LDS buffer.
// Transform axis has stride S; the 256 untouched groups are tiled 16 at a
// time into the B operand (16x16), multiplied by the constant H16 via four
// chained K=4 f32 WMMAs, and written back in place.
//   read/write index = blk*blkStride + n*nStride + k*S
__device__ __forceinline__ void whvi_fwht_stage(float* buf, int lane,
                                                int S, int blkStride, int nStride) {
    const int n    = lane & 15;   // column of B / C, also M-row of A fragment
    const int half = lane >> 4;   // lane half selects K pair / C row half

    #pragma unroll 1
    for (int blk = 0; blk < 16; ++blk) {
        const int base = blk * blkStride + n * nStride;
        v8f c = {};
        #pragma unroll
        for (int k0 = 0; k0 < 16; k0 += 4) {
            const int kk = k0 + 2 * half;      // K index held by VGPR0 of this lane
            v2f a, b;
            a.x = whvi_h16(n, kk);             // A: H16 generated in-register
            a.y = whvi_h16(n, kk + 1);
            b.x = buf[base + kk * S];          // B: rows kk, kk+1 of the 16x16 tile
            b.y = buf[base + (kk + 1) * S];
            c = __builtin_amdgcn_wmma_f32_16x16x4_f32(
                    /*neg_a=*/false, a, /*neg_b=*/false, b,
                    /*c_mod=*/(short)0, c, /*reuse_a=*/false, /*reuse_b=*/false);
        }
        // C layout: VGPR r holds row m = r + 8*half, column n
        #pragma unroll
        for (int r = 0; r < 8; ++r) {
            const int m = r + 8 * half;
            buf[base + m * S] = c[r];
        }
    }
}

__device__ __forceinline__ void whvi_fwht4096(float* buf, int lane) {
    whvi_fwht_stage(buf, lane, /*S=*/256, /*blkStride=*/16,  /*nStride=*/1);
    whvi_fwht_stage(buf, lane, /*S=*/16,  /*blkStride=*/256, /*nStride=*/1);
    whvi_fwht_stage(buf, lane, /*S=*/1,   /*blkStride=*/256, /*nStride=*/16);
}

// Prologue: u = g_mu + softplus(g_rho) * epsilon  (stable softplus)
__global__ void whvi_u_kernel(const float* __restrict__ g_mu,
                              const float* __restrict__ g_rho,
                              const float* __restrict__ eps,
                              float* __restrict__ u) {
    const int i = blockIdx.x * blockDim.x + threadIdx.x;
    if (i < WHVI_D) {
        const float r  = g_rho[i];
        const float sp = fmaxf(r, 0.0f) + log1pf(expf(-fabsf(r)));
        u[i] = g_mu[i] + sp * eps[i];
    }
}

__global__ __launch_bounds__(32 * WHVI_WAVES)
void whvi_fwht_kernel(const float* __restrict__ x,
                      const float* __restrict__ s1,
                      const float* __restrict__ s2,
                      const float* __restrict__ u,
                      float* __restrict__ out, int nrows) {
    __shared__ float lds[WHVI_WAVES][WHVI_D];   // 64 KB: one 16 KB row per wave

    const int wave = threadIdx.x >> 5;
    const int lane = threadIdx.x & 31;
    const int row  = blockIdx.x * WHVI_WAVES + wave;
    if (row >= nrows) return;                    // wave-uniform exit

    float* buf = lds[wave];

    // ---- load row (coalesced float4) and scale by s2 ----
    const float4* xr  = (const float4*)(x + (size_t)row * WHVI_D);
    const float4* s2v = (const float4*)s2;
    float4*       bv  = (float4*)buf;
    #pragma unroll 1
    for (int c = 0; c < WHVI_D / 4 / 32; ++c) {
        const int i4 = c * 32 + lane;
        const float4 v = xr[i4];
        const float4 s = s2v[i4];
        float4 t; t.x = v.x * s.x; t.y = v.y * s.y; t.z = v.z * s.z; t.w = v.w * s.w;
        bv[i4] = t;
    }

    // ---- first FWHT ----
    whvi_fwht4096(buf, lane);

    // ---- elementwise multiply by u ----
    const float4* uv = (const float4*)u;
    #pragma unroll 1
    for (int c = 0; c < WHVI_D / 4 / 32; ++c) {
        const int i4 = c * 32 + lane;
        float4 t = bv[i4];
        const float4 s = uv[i4];
        t.x *= s.x; t.y *= s.y; t.z *= s.z; t.w *= s.w;
        bv[i4] = t;
    }

    // ---- second FWHT ----
    whvi_fwht4096(buf, lane);

    // ---- scale by s1 and store (coalesced float4) ----
    const float4* s1v = (const float4*)s1;
    float4*       yr  = (float4*)(out + (size_t)row * WHVI_D);
    #pragma unroll 1
    for (int c = 0; c < WHVI_D / 4 / 32; ++c) {
        const int i4 = c * 32 + lane;
        float4 t = bv[i4];
        const float4 s = s1v[i4];
        t.x *= s.x; t.y *= s.y; t.z *= s.z; t.w *= s.w;
        yr[i4] = t;
    }
}

extern "C" void kernel_launch(void* const* d_in, const int* in_sizes, int n_in,
                              void* d_out, int out_size, void* d_ws, size_t ws_size,
                              hipStream_t stream) {
    const float* x     = (const float*)d_in[0];
    const float* s1    = (const float*)d_in[1];
    const float* s2    = (const float*)d_in[2];
    const float* g_mu  = (const float*)d_in[3];
    const float* g_rho = (const float*)d_in[4];
    const float* eps   = (const float*)d_in[5];
    // d_in[6] is H: unused — the Hadamard transform is computed directly.

    float* u   = (float*)d_ws;          // 4096 floats of scratch
    float* out = (float*)d_out;

    const int nrows = in_sizes[0] / WHVI_D;   // 8192

    whvi_u_kernel<<<(WHVI_D + 255) / 256, 256, 0, stream>>>(g_mu, g_rho, eps, u);

    const int blocks = (nrows + WHVI_WAVES - 1) / WHVI_WAVES;
    whvi_fwht_kernel<<<blocks, 32 * WHVI_WAVES, 0, stream>>>(x, s1, s2, u, out, nrows);
}